// LorentzRankingLoss_20091857011219
// MI455X (gfx1250) — compile-verified
//
#include <hip/hip_runtime.h>
#include <math.h>

// Problem constants from the reference
#define DD      32
#define HWZ     (128*128*128)
#define NC      105
#define NCPAD   112          // 7 tiles of 16
#define NTN     7            // N tiles
#define NNEG    8
#define MARGINF 0.1f
#define GUARDF  (1.0f + 1e-7f)

typedef __attribute__((ext_vector_type(2))) float v2f;
typedef __attribute__((ext_vector_type(8))) float v8f;

// ---------------------------------------------------------------------------
// Kernel 1: per-class Lorentz time component ct[c] = sqrt(1 + |label_c|^2)
// ---------------------------------------------------------------------------
__global__ void lrl_label_time(const float* __restrict__ label, float* __restrict__ ct) {
    int c = blockIdx.x * blockDim.x + threadIdx.x;
    if (c < NC) {
        float s = 0.f;
#pragma unroll
        for (int d = 0; d < DD; ++d) { float v = label[c * DD + d]; s += v * v; }
        ct[c] = sqrtf(1.0f + s);
    }
}

// ---------------------------------------------------------------------------
// Kernel 2: S[k, c] = anchor_k . label_c via V_WMMA_F32_16X16X4_F32.
// One wave per 16-row M-tile; the wave loops over all 7 N-tiles so each
// A fragment (expensive 8MB-strided gather) is loaded once and reused for
// 7 WMMAs. Pad columns (n >= 105) use a clamped row index and produce
// garbage that is never read -> no predication, no exec-mask branching.
//   A frag: lane = m + 16*(k>>1), vgpr = k&1
//   B frag: lane = n + 16*(k>>1), vgpr = k&1   (label rows, b64 loads)
//   C/D:    VGPR r -> M = r + 8*(lane>>4), N = lane&15
// Also accumulates |anchor|^2 in-flight (lanes L and L+16 jointly cover all
// 32 dims of row L&15; combined with one shfl_xor) -> asq[] for the loss.
// ---------------------------------------------------------------------------
__global__ void lrl_gemm_wmma(const float* __restrict__ voxel,
                              const float* __restrict__ label,
                              const int*   __restrict__ sidx,
                              float*       __restrict__ S,
                              float*       __restrict__ asq,
                              int K) {
    const int tm     = blockIdx.x;
    const int lane   = threadIdx.x;     // 0..31
    const int half   = lane >> 4;       // upper K half select
    const int lane15 = lane & 15;

    const int m   = tm * 16 + lane15;
    const int idx = sidx[m < K ? m : (K - 1)];   // clamp; padded rows never read
    const long long vb = (long long)idx;

    // Clamped label-row pointers for each of the 7 N-tiles this wave owns.
    const float* lrow[NTN];
#pragma unroll
    for (int tn = 0; tn < NTN; ++tn) {
        int n = tn * 16 + lane15;
        lrow[tn] = label + (long long)(n < NC ? n : (NC - 1)) * DD;
    }

    v8f acc[NTN];
#pragma unroll
    for (int tn = 0; tn < NTN; ++tn)
#pragma unroll
        for (int r = 0; r < 8; ++r) acc[tn][r] = 0.f;

    float sq = 0.f;   // partial |anchor|^2 over this lane's K slots

#pragma unroll
    for (int kk = 0; kk < 8; ++kk) {
        const int k0 = 4 * kk + 2 * half;
        v2f a;
        a.x = voxel[(long long)k0 * HWZ + vb];
        a.y = voxel[(long long)(k0 + 1) * HWZ + vb];
        sq += a.x * a.x + a.y * a.y;
#pragma unroll
        for (int tn = 0; tn < NTN; ++tn) {
            const v2f b = *reinterpret_cast<const v2f*>(lrow[tn] + k0);
            // (neg_a, A, neg_b, B, c_mod, C, reuse_a, reuse_b)
            acc[tn] = __builtin_amdgcn_wmma_f32_16x16x4_f32(false, a, false, b,
                                                            (short)0, acc[tn],
                                                            false, false);
        }
    }

    // Combine the two K-halves: lanes L and L+16 hold complementary dims of m.
    sq += __shfl_xor(sq, 16, 32);
    if (lane < 16 && m < K) asq[m] = sq;

#pragma unroll
    for (int tn = 0; tn < NTN; ++tn) {
        const int nn = tn * 16 + lane15;
#pragma unroll
        for (int r = 0; r < 8; ++r) {
            const int mm = tm * 16 + r + 8 * half;     // < Kpad by construction
            S[(long long)mm * NCPAD + nn] = acc[tn][r];
        }
    }
}

// ---------------------------------------------------------------------------
// Kernel 3: triplet loss from precomputed scores; single-workgroup reduction.
// ---------------------------------------------------------------------------
__global__ void lrl_loss(const int*   __restrict__ scls,
                         const int*   __restrict__ negc,
                         const float* __restrict__ ct,
                         const float* __restrict__ asq,
                         const float* __restrict__ S,
                         float*       __restrict__ out,
                         int K) {
    __shared__ float red[1024];
    const int tid = threadIdx.x;
    float acc = 0.f;
    for (int k = tid; k < K; k += (int)blockDim.x) {
        const float xt = sqrtf(1.0f + asq[k]);
        const float* Srow = S + (long long)k * NCPAD;
        const int p = scls[k];
        const float dpos = acoshf(fmaxf(xt * ct[p] - Srow[p], GUARDF));
        float t = 0.f;
#pragma unroll
        for (int j = 0; j < NNEG; ++j) {
            const int cn = negc[k * NNEG + j];
            const float dneg = acoshf(fmaxf(xt * ct[cn] - Srow[cn], GUARDF));
            t += fmaxf(MARGINF + dpos - dneg, 0.f);
        }
        acc += t * (1.0f / NNEG);
    }
    red[tid] = acc;
    __syncthreads();
    for (int s = (int)blockDim.x >> 1; s > 0; s >>= 1) {
        if (tid < s) red[tid] += red[tid + s];
        __syncthreads();
    }
    if (tid == 0) out[0] = (K > 0) ? red[0] / (float)K : 0.f;
}

// ---------------------------------------------------------------------------
// Fallback (no workspace needed): direct per-anchor evaluation.
// ---------------------------------------------------------------------------
__global__ void lrl_loss_direct(const float* __restrict__ voxel,
                                const float* __restrict__ label,
                                const int*   __restrict__ sidx,
                                const int*   __restrict__ scls,
                                const int*   __restrict__ negc,
                                float*       __restrict__ out,
                                int K) {
    __shared__ float red[1024];
    const int tid = threadIdx.x;
    float acc = 0.f;
    for (int k = tid; k < K; k += (int)blockDim.x) {
        const int idx = sidx[k];
        float a[DD];
        float s2 = 0.f;
#pragma unroll
        for (int d = 0; d < DD; ++d) {
            a[d] = voxel[(long long)d * HWZ + idx];
            s2 += a[d] * a[d];
        }
        const float xt = sqrtf(1.0f + s2);
        const int p = scls[k];
        float dotp = 0.f, n2p = 0.f;
#pragma unroll
        for (int d = 0; d < DD; ++d) {
            const float lv = label[p * DD + d];
            dotp += a[d] * lv; n2p += lv * lv;
        }
        const float dpos = acoshf(fmaxf(xt * sqrtf(1.0f + n2p) - dotp, GUARDF));
        float t = 0.f;
        for (int j = 0; j < NNEG; ++j) {
            const int cn = negc[k * NNEG + j];
            float dotn = 0.f, n2n = 0.f;
#pragma unroll
            for (int d = 0; d < DD; ++d) {
                const float lv = label[cn * DD + d];
                dotn += a[d] * lv; n2n += lv * lv;
            }
            const float dneg = acoshf(fmaxf(xt * sqrtf(1.0f + n2n) - dotn, GUARDF));
            t += fmaxf(MARGINF + dpos - dneg, 0.f);
        }
        acc += t * (1.0f / NNEG);
    }
    red[tid] = acc;
    __syncthreads();
    for (int s = (int)blockDim.x >> 1; s > 0; s >>= 1) {
        if (tid < s) red[tid] += red[tid + s];
        __syncthreads();
    }
    if (tid == 0) out[0] = (K > 0) ? red[0] / (float)K : 0.f;
}

extern "C" void kernel_launch(void* const* d_in, const int* in_sizes, int n_in,
                              void* d_out, int out_size, void* d_ws, size_t ws_size,
                              hipStream_t stream) {
    const float* voxel = (const float*)d_in[0];
    // d_in[1] = labels (unused by the loss itself)
    const float* label = (const float*)d_in[2];
    const int*   sidx  = (const int*)d_in[3];
    const int*   scls  = (const int*)d_in[4];
    const int*   negc  = (const int*)d_in[5];
    float*       out   = (float*)d_out;

    const int    K      = in_sizes[3];
    const int    mtiles = (K + 15) / 16;
    const size_t kpad   = (size_t)mtiles * 16;
    const size_t need   = (NCPAD + kpad + kpad * NCPAD) * sizeof(float);

    if (d_ws != nullptr && ws_size >= need) {
        float* ct  = (float*)d_ws;
        float* asq = ct + NCPAD;
        float* S   = asq + kpad;
        lrl_label_time<<<1, 128, 0, stream>>>(label, ct);
        lrl_gemm_wmma<<<mtiles, 32, 0, stream>>>(voxel, label, sidx, S, asq, K);
        lrl_loss<<<1, 1024, 0, stream>>>(scls, negc, ct, asq, S, out, K);
    } else {
        lrl_loss_direct<<<1, 1024, 0, stream>>>(voxel, label, sidx, scls, negc, out, K);
    }
}